// GuidanceNodeGenerator_60576218743376
// MI455X (gfx1250) — compile-verified
//
#include <hip/hip_runtime.h>
#include <math.h>

typedef __attribute__((ext_vector_type(2))) float v2f;
typedef __attribute__((ext_vector_type(8))) float v8f;

#define HID 64
#define INCH 128

// ---------------- elementwise / setup kernels ----------------

__global__ __launch_bounds__(256) void k_fill(float* p, float v, int n) {
    int i = blockIdx.x * 256 + threadIdx.x;
    if (i < n) p[i] = v;
}

__global__ __launch_bounds__(256) void k_deg_edges(const int* __restrict__ dst,
                                                   float* deg, int E) {
    int e = blockIdx.x * 256 + threadIdx.x;
    if (e < E) atomicAdd(&deg[dst[e]], 1.0f);
}

__global__ __launch_bounds__(256) void k_rsqrt(const float* __restrict__ deg,
                                               float* dis, int n) {
    int i = blockIdx.x * 256 + threadIdx.x;
    if (i < n) dis[i] = rsqrtf(deg[i]);   // deg >= 1 (self-loops)
}

__global__ __launch_bounds__(256) void k_mask_scatter(const int* __restrict__ bn,
                                                      float* mask, int B) {
    int i = blockIdx.x * 256 + threadIdx.x;
    if (i < B) mask[bn[i]] = 0.0f;
}

// ---------------- WMMA f32 GEMM: C[n,64] = (rowmask .* A[n,K]) @ B[K,64] (+bias)(relu) ----
// One wave computes one 16x16 f32 tile with V_WMMA_F32_16X16X4_F32.
// Block = 256 threads = 8 waves = 2 row-tiles x 4 col-tiles -> 32 rows x 64 cols per block.

__global__ __launch_bounds__(256) void k_gemm_wmma_f32(
    const float* __restrict__ A, const float* __restrict__ Bm,
    float* __restrict__ C, const float* __restrict__ mask,
    const float* __restrict__ bias, int n, int K, int applyRelu)
{
    const int ldb = HID;
    int wave = threadIdx.x >> 5;
    int lane = threadIdx.x & 31;
    int rowBase = (blockIdx.x * 2 + (wave >> 2)) * 16;
    int colBase = (wave & 3) * 16;
    int half = lane >> 4;            // 0: lanes 0-15, 1: lanes 16-31
    int l16  = lane & 15;

    int row = rowBase + l16;         // A-fragment row (M = lane&15)
    int rclamp = row < n ? row : 0;
    float scaleA = (row < n) ? 1.0f : 0.0f;
    if (mask) scaleA *= mask[rclamp];
    int col = colBase + l16;         // B/C/D column (N = lane&15)

    const float* Arow = A + (size_t)rclamp * (size_t)K;

    v8f acc = {0.f,0.f,0.f,0.f,0.f,0.f,0.f,0.f};
    for (int k0 = 0; k0 < K; k0 += 4) {
        int ka = k0 + 2 * half;      // this lane holds K = ka, ka+1
        v2f a, b;
        a.x = Arow[ka]     * scaleA;
        a.y = Arow[ka + 1] * scaleA;
        b.x = Bm[(size_t)ka       * ldb + col];
        b.y = Bm[(size_t)(ka + 1) * ldb + col];
        acc = __builtin_amdgcn_wmma_f32_16x16x4_f32(
                  false, a, false, b, (short)0, acc, false, false);
    }

    float bv = bias ? bias[col] : 0.0f;
    #pragma unroll
    for (int v = 0; v < 8; ++v) {
        int rr = rowBase + v + 8 * half;   // D: M = vgpr + 8*half
        if (rr < n) {
            float x = acc[v] + bv;
            if (applyRelu) x = x > 0.0f ? x : 0.0f;
            C[(size_t)rr * HID + col] = x;
        }
    }
}

// ---------------- aggregation (L2-resident scatter-add) ----------------

__global__ __launch_bounds__(256) void k_agg_init(const float* __restrict__ lin,
                                                  const float* __restrict__ dis,
                                                  float* agg, int n) {
    int i = blockIdx.x * 256 + threadIdx.x;
    if (i < n * HID) {
        float d = dis[i >> 6];               // self-loop: norm = dis[n]^2
        agg[i] = lin[i] * d * d;
    }
}

__global__ __launch_bounds__(256) void k_agg_edges(
    const float* __restrict__ lin, const int* __restrict__ src,
    const int* __restrict__ dst, const float* __restrict__ dis,
    float* agg, int E)
{
    int e = blockIdx.x * 8 + (threadIdx.x >> 5);   // one wave per edge
    if (e >= E) return;
    int lane = threadIdx.x & 31;
    int s = src[e], d = dst[e];
    float nrm = dis[s] * dis[d];
    const float2* hp = (const float2*)(lin + (size_t)s * HID);
    float2 v = hp[lane];                           // 2 channels / lane
    float* op = agg + (size_t)d * HID + lane * 2;
    atomicAdd(op,     v.x * nrm);
    atomicAdd(op + 1, v.y * nrm);
}

__global__ __launch_bounds__(256) void k_bias_relu(float* agg, const float* __restrict__ b,
                                                   int n, int relu) {
    int i = blockIdx.x * 256 + threadIdx.x;
    if (i < n * HID) {
        float v = agg[i] + b[i & (HID - 1)];
        if (relu) v = v > 0.0f ? v : 0.0f;
        agg[i] = v;
    }
}

// ---------------- loss tail ----------------
// sm layout: [0:64) pos_sum, [64:128) neg_sum, [128] cnt_pos, [129] cnt_neg,
//            [130:194) vpos, [194:258) vneg, [258] ctx_acc, [259] loc_acc

__global__ __launch_bounds__(64) void k_gather_bf(const float* __restrict__ g,
                                                  const int* __restrict__ bn,
                                                  const int* __restrict__ labels,
                                                  float* bf, float* bly) {
    int i = blockIdx.x, c = threadIdx.x;
    int node = bn[i];
    bf[i * HID + c] = g[(size_t)node * HID + c];
    if (c == 0) bly[i] = (float)labels[node];
}

__global__ __launch_bounds__(64) void k_protos(const float* __restrict__ bf,
                                               const float* __restrict__ bly,
                                               float* sm) {
    int i = blockIdx.x, c = threadIdx.x;
    float y = bly[i];
    float v = bf[i * HID + c];
    atomicAdd(&sm[c],      v * y);
    atomicAdd(&sm[64 + c], v * (1.0f - y));
    if (c == 0) { atomicAdd(&sm[128], y); atomicAdd(&sm[129], 1.0f - y); }
}

__global__ __launch_bounds__(64) void k_proto_div(float* sm) {
    int c = threadIdx.x;
    float cp = sm[128] > 1.0f ? sm[128] : 1.0f;
    float cn = sm[129] > 1.0f ? sm[129] : 1.0f;
    sm[c]      = sm[c]      / cp;
    sm[64 + c] = sm[64 + c] / cn;
}

__global__ __launch_bounds__(64) void k_ctx_vec(const float* __restrict__ Wctx, float* sm) {
    int d = threadIdx.x;
    float vp = 0.f, vn = 0.f;
    for (int e = 0; e < HID; ++e) {
        float w = Wctx[d * HID + e];
        vp += w * sm[e];
        vn += w * sm[64 + e];
    }
    sm[130 + d] = vp;
    sm[194 + d] = vn;
}

__device__ __forceinline__ float bce_logit(float z, float y) {
    return fmaxf(z, 0.0f) - z * y + log1pf(expf(-fabsf(z)));
}

__global__ __launch_bounds__(256) void k_ctx_loss(const float* __restrict__ bf,
                                                  const float* __restrict__ bly,
                                                  float* sm, int B) {
    int i = blockIdx.x * 256 + threadIdx.x;
    if (i >= B) return;
    const float* r = bf + i * HID;
    float ps = 0.f, ns = 0.f;
    for (int d = 0; d < HID; ++d) { ps += r[d] * sm[130 + d]; ns += r[d] * sm[194 + d]; }
    float z = ps - ns;                       // (+bctx) - (+bctx) cancels
    atomicAdd(&sm[258], bce_logit(z, bly[i]));
}

__global__ __launch_bounds__(256) void k_local_loss(const float* __restrict__ bfW,
                                                    const float* __restrict__ env,
                                                    const int* __restrict__ bn,
                                                    const float* __restrict__ bly,
                                                    const float* __restrict__ bloc,
                                                    float* sm, int B) {
    int i = blockIdx.x * 256 + threadIdx.x;
    if (i >= B) return;
    const float* r = bfW + i * HID;
    const float* e = env + (size_t)bn[i] * HID;
    float s = 0.f;
    for (int d = 0; d < HID; ++d) s += r[d] * e[d];
    s += bloc[0];
    atomicAdd(&sm[259], bce_logit(s, 1.0f - bly[i]));
}

__global__ void k_final(const float* __restrict__ sm, float* out, float invB) {
    out[0] = 0.5f * (sm[258] * invB) + 0.5f * (sm[259] * invB);
}

// ---------------- host-side launch ----------------

extern "C" void kernel_launch(void* const* d_in, const int* in_sizes, int n_in,
                              void* d_out, int out_size, void* d_ws, size_t ws_size,
                              hipStream_t stream) {
    const float* x      = (const float*)d_in[0];
    const int*   eidx   = (const int*)d_in[1];
    const int*   bn     = (const int*)d_in[2];
    const int*   labels = (const int*)d_in[3];
    const float* W1  = (const float*)d_in[4];
    const float* b1  = (const float*)d_in[5];
    const float* W2  = (const float*)d_in[6];
    const float* b2  = (const float*)d_in[7];
    const float* Wp  = (const float*)d_in[8];
    const float* bp  = (const float*)d_in[9];
    const float* Wctx = (const float*)d_in[10];
    // bctx (d_in[11]) cancels in pos_sim - neg_sim
    const float* Wloc = (const float*)d_in[12];
    const float* bloc = (const float*)d_in[13];

    const int N = in_sizes[0] / INCH;
    const int E = in_sizes[1] / 2;
    const int B = in_sizes[2];
    const int* src = eidx;         // edge_index[0]
    const int* dst = eidx + E;     // edge_index[1]

    float* ws   = (float*)d_ws;
    float* deg  = ws;                         // N
    float* dis  = deg + N;                    // N
    float* maskv= dis + N;                    // N
    float* lin  = maskv + N;                  // N*64
    float* agg  = lin + (size_t)N * HID;      // N*64
    float* env  = agg + (size_t)N * HID;      // N*64
    float* bf   = env + (size_t)N * HID;      // B*64
    float* bfW  = bf + (size_t)B * HID;       // B*64
    float* bly  = bfW + (size_t)B * HID;      // B
    float* sm   = bly + B;                    // 260 floats

    float* guidance = (float*)d_out;          // N*64
    float* loss_out = guidance + (size_t)N * HID;

    const int gN   = (N + 255) / 256;
    const int gNH  = (N * HID + 255) / 256;
    const int gE   = (E + 255) / 256;
    const int gEw  = (E + 7) / 8;
    const int gB   = (B + 255) / 256;
    const int gGN  = (N + 31) / 32;
    const int gGB  = (B + 31) / 32;

    // degree + norm
    k_fill<<<gN, 256, 0, stream>>>(deg, 1.0f, N);                 // self-loop
    k_deg_edges<<<gE, 256, 0, stream>>>(dst, deg, E);
    k_rsqrt<<<gN, 256, 0, stream>>>(deg, dis, N);
    // mask for env encode
    k_fill<<<gN, 256, 0, stream>>>(maskv, 1.0f, N);
    k_mask_scatter<<<gB, 256, 0, stream>>>(bn, maskv, B);

    // ---- encode(x) -> guidance ----
    k_gemm_wmma_f32<<<gGN, 256, 0, stream>>>(x, W1, lin, nullptr, nullptr, N, INCH, 0);
    k_agg_init<<<gNH, 256, 0, stream>>>(lin, dis, agg, N);
    k_agg_edges<<<gEw, 256, 0, stream>>>(lin, src, dst, dis, agg, E);
    k_bias_relu<<<gNH, 256, 0, stream>>>(agg, b1, N, 1);
    k_gemm_wmma_f32<<<gGN, 256, 0, stream>>>(agg, W2, lin, nullptr, nullptr, N, HID, 0);
    k_agg_init<<<gNH, 256, 0, stream>>>(lin, dis, agg, N);
    k_agg_edges<<<gEw, 256, 0, stream>>>(lin, src, dst, dis, agg, E);
    k_bias_relu<<<gNH, 256, 0, stream>>>(agg, b2, N, 1);
    k_gemm_wmma_f32<<<gGN, 256, 0, stream>>>(agg, Wp, guidance, nullptr, bp, N, HID, 0);

    // ---- contrastive (context) loss ----
    k_fill<<<2, 256, 0, stream>>>(sm, 0.0f, 260);
    k_gather_bf<<<B, 64, 0, stream>>>(guidance, bn, labels, bf, bly);
    k_protos<<<B, 64, 0, stream>>>(bf, bly, sm);
    k_proto_div<<<1, 64, 0, stream>>>(sm);
    k_ctx_vec<<<1, 64, 0, stream>>>(Wctx, sm);
    k_ctx_loss<<<gB, 256, 0, stream>>>(bf, bly, sm, B);

    // ---- encode(masked x) -> env ----
    k_gemm_wmma_f32<<<gGN, 256, 0, stream>>>(x, W1, lin, maskv, nullptr, N, INCH, 0);
    k_agg_init<<<gNH, 256, 0, stream>>>(lin, dis, agg, N);
    k_agg_edges<<<gEw, 256, 0, stream>>>(lin, src, dst, dis, agg, E);
    k_bias_relu<<<gNH, 256, 0, stream>>>(agg, b1, N, 1);
    k_gemm_wmma_f32<<<gGN, 256, 0, stream>>>(agg, W2, lin, nullptr, nullptr, N, HID, 0);
    k_agg_init<<<gNH, 256, 0, stream>>>(lin, dis, agg, N);
    k_agg_edges<<<gEw, 256, 0, stream>>>(lin, src, dst, dis, agg, E);
    k_bias_relu<<<gNH, 256, 0, stream>>>(agg, b2, N, 1);
    k_gemm_wmma_f32<<<gGN, 256, 0, stream>>>(agg, Wp, env, nullptr, bp, N, HID, 0);

    // ---- local loss ----
    k_gemm_wmma_f32<<<gGB, 256, 0, stream>>>(bf, Wloc, bfW, nullptr, nullptr, B, HID, 0);
    k_local_loss<<<gB, 256, 0, stream>>>(bfW, env, bn, bly, bloc, sm, B);
    k_final<<<1, 1, 0, stream>>>(sm, loss_out, 1.0f / (float)B);
}